// ConceptContrastiveLoss_79207786873536
// MI455X (gfx1250) — compile-verified
//
#include <hip/hip_runtime.h>
#include <hip/hip_bf16.h>
#include <math.h>

// ---------------------------------------------------------------------------
// ConceptContrastiveLoss for MI455X (gfx1250)
//
// expert:   [B=256, S=2048, D=128] f32
// violator: [B=256, S=2048, D=128] f32
//
// Phase 1 (bandwidth-bound, ~22us at 23.3 TB/s): centroid mean over S,
//          coalesced float4 streaming loads, plus squared norms.
// Phase 2 (tiny): three 256x256 Gram matrices over D=128 via
//          V_WMMA_F32_16X16X4_F32 (f32 MACs -> matches f32 reference),
//          fused with the per-pair loss math and a wave reduction.
// Phase 3: scalar combine.
// ---------------------------------------------------------------------------

typedef __attribute__((ext_vector_type(2))) float v2f;
typedef __attribute__((ext_vector_type(8))) float v8f;

#define BATCH 256
#define SEQ   2048
#define DIM   128

#define MARGIN 10.0f
#define ALPHA  3.0f
#define BETA   0.3f
#define GAMMA  0.3f

// ---------------------------------------------------------------------------
// Kernel 1: centroids + squared norms.
// One block per (tensor, batch) pair. 256 threads: 32 float4-lanes per row
// (128 floats), 8 row-phases. Every iteration the block reads 8 full
// contiguous rows (4 KB) with global_load_b128 -> perfectly coalesced.
// ---------------------------------------------------------------------------
__global__ __launch_bounds__(256)
void centroid_kernel(const float* __restrict__ expert,
                     const float* __restrict__ violator,
                     float* __restrict__ cent_e,   // [256*128]
                     float* __restrict__ cent_v,   // [256*128]
                     float* __restrict__ norm_e,   // [256]
                     float* __restrict__ norm_v)   // [256]
{
    const int b     = blockIdx.x & (BATCH - 1);
    const int which = blockIdx.x >> 8;           // 0 = expert, 1 = violator

    const float* __restrict__ src = which ? violator : expert;
    float* __restrict__ cent      = which ? cent_v : cent_e;
    float* __restrict__ nrm       = which ? norm_v : norm_e;

    const int tid = threadIdx.x;
    const int d4  = tid & 31;    // which float4 within the 128-float row
    const int ph  = tid >> 5;    // row phase 0..7

    const float4* __restrict__ base =
        (const float4*)(src + (size_t)b * SEQ * DIM);

    float4 acc = make_float4(0.f, 0.f, 0.f, 0.f);
    for (int s = ph; s < SEQ; s += 8) {
        float4 x = base[(size_t)s * 32 + d4];
        acc.x += x.x; acc.y += x.y; acc.z += x.z; acc.w += x.w;
    }

    __shared__ float sm[8 * DIM];   // [phase][d]
    sm[ph * DIM + d4 * 4 + 0] = acc.x;
    sm[ph * DIM + d4 * 4 + 1] = acc.y;
    sm[ph * DIM + d4 * 4 + 2] = acc.z;
    sm[ph * DIM + d4 * 4 + 3] = acc.w;
    __syncthreads();

    __shared__ float sq[DIM];
    if (tid < DIM) {
        float s = 0.f;
        #pragma unroll
        for (int p = 0; p < 8; ++p) s += sm[p * DIM + tid];
        float c = s * (1.0f / (float)SEQ);
        cent[(size_t)b * DIM + tid] = c;
        sq[tid] = c * c;
    }
    __syncthreads();
    for (int off = 64; off > 0; off >>= 1) {
        if (tid < off) sq[tid] += sq[tid + off];
        __syncthreads();
    }
    if (tid == 0) nrm[b] = sq[0];
}

// ---------------------------------------------------------------------------
// Kernel 2: zero the 3 loss accumulators (d_ws is poisoned by the harness).
// ---------------------------------------------------------------------------
__global__ void zero_partials_kernel(float* __restrict__ partials)
{
    if (threadIdx.x < 3) partials[threadIdx.x] = 0.0f;
}

// ---------------------------------------------------------------------------
// Kernel 3: WMMA Gram tiles + fused loss.
// grid = (16, 16, 3), block = 32 (one wave, EXEC all-ones as WMMA requires).
//   mode 0: sep  = sum over (i,j) of max(MARGIN - dist(E_i, V_j), 0)^2
//   mode 1: ee   = sum over i!=j of clamped sqdist(E_i, E_j)
//   mode 2: vv   = sum over i!=j of clamped sqdist(V_i, V_j)
//
// A-matrix 16x4 f32 layout (ISA 7.12.2): lanes 0-15 hold M=0..15 with K=0
// (vgpr0) / K=1 (vgpr1); lanes 16-31 hold K=2/K=3. B 4x16 mirrors with N
// across lanes. Since B here is centroid^T, both operands load the same
// row-major float2: row (tile*16 + lane&15), cols (4*kb + 2*(lane>>4)) .. +1.
// ---------------------------------------------------------------------------
__global__ __launch_bounds__(32)
void gram_loss_kernel(const float* __restrict__ cent_e,
                      const float* __restrict__ cent_v,
                      const float* __restrict__ norm_e,
                      const float* __restrict__ norm_v,
                      float* __restrict__ partials)
{
    const int mode = blockIdx.z;
    const int ti   = blockIdx.x;   // row tile of A
    const int tj   = blockIdx.y;   // row tile of B (columns of the Gram)

    const float* A;  const float* B;
    const float* nA; const float* nB;
    if (mode == 0)      { A = cent_e; B = cent_v; nA = norm_e; nB = norm_v; }
    else if (mode == 1) { A = cent_e; B = cent_e; nA = norm_e; nB = norm_e; }
    else                { A = cent_v; B = cent_v; nA = norm_v; nB = norm_v; }

    const int lane = threadIdx.x;
    const int half = lane >> 4;     // 0: K=0,1  1: K=2,3
    const int idx  = lane & 15;     // M for A-lanes, N for B-lanes

    const float* __restrict__ arow = A + (size_t)(ti * 16 + idx) * DIM + 2 * half;
    const float* __restrict__ brow = B + (size_t)(tj * 16 + idx) * DIM + 2 * half;

    v8f acc = {};
    #pragma unroll
    for (int kb = 0; kb < DIM / 4; ++kb) {
        v2f a = *(const v2f*)(arow + kb * 4);
        v2f b = *(const v2f*)(brow + kb * 4);
        // D = A x B + C ; 16x16x4 f32 (exact f32 MACs, matches reference)
        acc = __builtin_amdgcn_wmma_f32_16x16x4_f32(
            /*neg_a=*/false, a, /*neg_b=*/false, b,
            /*c_mod=*/(short)0, acc, /*reuse_a=*/false, /*reuse_b=*/false);
    }

    // C/D layout: vgpr v, lanes 0-15 -> M=v, lanes 16-31 -> M=8+v; N=lane&15.
    const int j   = tj * 16 + idx;
    const float nbj = nB[j];

    float local = 0.0f;
    #pragma unroll
    for (int v = 0; v < 8; ++v) {
        const int i = ti * 16 + 8 * half + v;
        float sq = nA[i] + nbj - 2.0f * acc[v];
        sq = fmaxf(sq, 0.0f);
        if (mode == 0) {
            float d = sqrtf(sq + 1e-12f);
            float t = fmaxf(MARGIN - d, 0.0f);
            local += t * t;
        } else {
            local += (i == j) ? 0.0f : sq;
        }
    }

    // wave32 reduction
    #pragma unroll
    for (int off = 16; off > 0; off >>= 1)
        local += __shfl_xor(local, off, 32);

    if (lane == 0) atomicAdd(&partials[mode], local);
}

// ---------------------------------------------------------------------------
// Kernel 4: combine partial sums into the scalar loss.
// ---------------------------------------------------------------------------
__global__ void finalize_kernel(const float* __restrict__ partials,
                                float* __restrict__ out)
{
    if (threadIdx.x == 0) {
        const float n_pairs_ev = (float)(BATCH * BATCH);              // 65536
        const float n_pairs_tri = (float)(BATCH * (BATCH - 1) / 2);   // 32640
        float sep = partials[0] / n_pairs_ev;
        float ee  = partials[1] * 0.5f / n_pairs_tri;  // off-diag sum -> triu sum
        float vv  = partials[2] * 0.5f / n_pairs_tri;
        out[0] = ALPHA * sep + BETA * ee + GAMMA * vv;
    }
}

// ---------------------------------------------------------------------------
// Launch
// ---------------------------------------------------------------------------
extern "C" void kernel_launch(void* const* d_in, const int* in_sizes, int n_in,
                              void* d_out, int out_size, void* d_ws, size_t ws_size,
                              hipStream_t stream)
{
    const float* expert   = (const float*)d_in[0];
    const float* violator = (const float*)d_in[1];
    float* out = (float*)d_out;

    // workspace layout (floats)
    float* ws      = (float*)d_ws;
    float* cent_e  = ws;                         // 256*128
    float* cent_v  = cent_e + BATCH * DIM;       // 256*128
    float* norm_e  = cent_v + BATCH * DIM;       // 256
    float* norm_v  = norm_e + BATCH;             // 256
    float* partial = norm_v + BATCH;             // 3

    centroid_kernel<<<dim3(2 * BATCH), dim3(256), 0, stream>>>(
        expert, violator, cent_e, cent_v, norm_e, norm_v);

    zero_partials_kernel<<<dim3(1), dim3(32), 0, stream>>>(partial);

    gram_loss_kernel<<<dim3(BATCH / 16, BATCH / 16, 3), dim3(32), 0, stream>>>(
        cent_e, cent_v, norm_e, norm_v, partial);

    finalize_kernel<<<dim3(1), dim3(32), 0, stream>>>(partial, out);
}